// HashGrid_28467043238537
// MI455X (gfx1250) — compile-verified
//
#include <hip/hip_runtime.h>
#include <math.h>

#define N_LEVELS 16
#define LOG2_T 19
#define TBL (1u << LOG2_T)
#define P1 2654435761u
#define P2 805459861u

// Level-0 dense table staged in LDS. res=16, but boundary corners reach
// idx = 16 + 16*16 + 256*16 = 4368, so stage 4384 entries (35 KB).
#define S0N 4384u

typedef int   v4i __attribute__((ext_vector_type(4)));
typedef float v4f __attribute__((ext_vector_type(4)));

struct LevelConsts {
  float scale[N_LEVELS];
  int   res[N_LEVELS];   // >0: dense resolution, 0: spatial hash
};

__global__ __launch_bounds__(256) void hashgrid_fwd(
    const float* __restrict__ x,
    const float2* __restrict__ emb,
    float* __restrict__ out,
    int n, LevelConsts c)
{
  __shared__ float2 s0[S0N];

  const int tid = threadIdx.x;

  // ---- cooperative fill of level-0 dense table into LDS ----
  {
    const int n4 = (int)(S0N / 2);               // 2192 16-byte chunks
#if __has_builtin(__builtin_amdgcn_global_load_async_to_lds_b128) && \
    __has_builtin(__builtin_amdgcn_s_wait_asynccnt)
    typedef __attribute__((address_space(1))) v4i* gv4_t;
    typedef __attribute__((address_space(3))) v4i* lv4_t;
    gv4_t gsrc = (gv4_t)emb;                     // level 0 starts at offset 0
    lv4_t ldst = (lv4_t)s0;
    for (int i = tid; i < n4; i += 256) {
      __builtin_amdgcn_global_load_async_to_lds_b128(gsrc + i, ldst + i, 0, 0);
    }
    __builtin_amdgcn_s_wait_asynccnt(0);
#else
    const float4* gsrc = (const float4*)emb;
    float4* ldst = (float4*)s0;
    for (int i = tid; i < n4; i += 256) ldst[i] = gsrc[i];
#endif
  }
  __syncthreads();

  // ---- grid-stride over points ----
  for (int p = blockIdx.x * 256 + tid; p < n; p += gridDim.x * 256) {
    const float x0 = (x[3 * p + 0] + 1.0f) * 0.5f;
    const float x1 = (x[3 * p + 1] + 1.0f) * 0.5f;
    const float x2 = (x[3 * p + 2] + 1.0f) * 0.5f;

    float of[2 * N_LEVELS];

#pragma unroll
    for (int l = 0; l < N_LEVELS; ++l) {
      const float sc = c.scale[l];
      const int   res = c.res[l];

      const float px = fmaf(x0, sc, 0.5f);
      const float py = fmaf(x1, sc, 0.5f);
      const float pz = fmaf(x2, sc, 0.5f);
      const float fx = floorf(px), fy = floorf(py), fz = floorf(pz);
      const float tx = px - fx, ty = py - fy, tz = pz - fz;
      const unsigned gx = (unsigned)(int)fx;
      const unsigned gy = (unsigned)(int)fy;
      const unsigned gz = (unsigned)(int)fz;

      unsigned idx[8];
      if (res > 0) {
        const unsigned r = (unsigned)res, r2 = r * r;
        const unsigned b = gx + r * gy + r2 * gz;
        idx[0] = b;          idx[1] = b + 1;
        idx[2] = b + r;      idx[3] = b + r + 1;
        idx[4] = b + r2;     idx[5] = b + r2 + 1;
        idx[6] = b + r2 + r; idx[7] = b + r2 + r + 1;
      } else {
        const unsigned hx0 = gx,             hx1 = gx + 1u;
        const unsigned hy0 = gy * P1,        hy1 = (gy + 1u) * P1;
        const unsigned hz0 = gz * P2,        hz1 = (gz + 1u) * P2;
        idx[0] = (hx0 ^ hy0 ^ hz0) & (TBL - 1u);
        idx[1] = (hx1 ^ hy0 ^ hz0) & (TBL - 1u);
        idx[2] = (hx0 ^ hy1 ^ hz0) & (TBL - 1u);
        idx[3] = (hx1 ^ hy1 ^ hz0) & (TBL - 1u);
        idx[4] = (hx0 ^ hy0 ^ hz1) & (TBL - 1u);
        idx[5] = (hx1 ^ hy0 ^ hz1) & (TBL - 1u);
        idx[6] = (hx0 ^ hy1 ^ hz1) & (TBL - 1u);
        idx[7] = (hx1 ^ hy1 ^ hz1) & (TBL - 1u);
      }

      // issue all 8 gathers before the reduction (MLP)
      float2 v[8];
      if (l == 0) {
#pragma unroll
        for (int k = 0; k < 8; ++k) v[k] = s0[idx[k]];
      } else {
        const float2* tbl = emb + (size_t)l * TBL;
#pragma unroll
        for (int k = 0; k < 8; ++k) v[k] = tbl[idx[k]];
      }

      const float wx[2] = {1.0f - tx, tx};
      const float wy[2] = {1.0f - ty, ty};
      const float wz[2] = {1.0f - tz, tz};
      float ax = 0.0f, ay = 0.0f;
#pragma unroll
      for (int k = 0; k < 8; ++k) {
        const float w = wx[k & 1] * wy[(k >> 1) & 1] * wz[(k >> 2) & 1];
        ax = fmaf(w, v[k].x, ax);
        ay = fmaf(w, v[k].y, ay);
      }
      of[2 * l]     = ax;
      of[2 * l + 1] = ay;
    }

    // stream the 128B result line with non-temporal b128 stores:
    // keep embeddings resident in L2, output goes straight to HBM.
    v4f* dst = (v4f*)(out + (size_t)p * (2 * N_LEVELS));
#pragma unroll
    for (int i = 0; i < 8; ++i) {
      v4f v4 = {of[4 * i], of[4 * i + 1], of[4 * i + 2], of[4 * i + 3]};
      __builtin_nontemporal_store(v4, dst + i);
    }
  }
}

static void fill_consts(LevelConsts* c) {
  const double s = exp(log(2048.0 / 16.0) / 15.0);   // PER_LEVEL_SCALE, double like numpy
  for (int l = 0; l < N_LEVELS; ++l) {
    const double sc = 16.0 * pow(s, (double)l) - 1.0;
    const int res = (int)ceil(sc) + 1;
    const long long r3 = (long long)res * res * res;
    c->scale[l] = (float)sc;
    c->res[l] = (r3 <= (long long)TBL) ? res : 0;
  }
}

extern "C" void kernel_launch(void* const* d_in, const int* in_sizes, int n_in,
                              void* d_out, int out_size, void* d_ws, size_t ws_size,
                              hipStream_t stream) {
  (void)n_in; (void)out_size; (void)d_ws; (void)ws_size;
  const float*  xin = (const float*)d_in[0];
  const float2* emb = (const float2*)d_in[1];
  float* out = (float*)d_out;
  const int n = in_sizes[0] / 3;

  LevelConsts c;
  fill_consts(&c);

  const int threads = 256;
  int blocks = (n + threads - 1) / threads;
  if (blocks > 4096) blocks = 4096;   // persistent-ish: amortize LDS fill
  hipLaunchKernelGGL(hashgrid_fwd, dim3(blocks), dim3(threads), 0, stream,
                     xin, emb, out, n, c);
}